// ConvolutionDownsample_50422916055266
// MI455X (gfx1250) — compile-verified
//
#include <hip/hip_runtime.h>
#include <hip/hip_bf16.h>

typedef __attribute__((ext_vector_type(2))) float v2f;
typedef __attribute__((ext_vector_type(8))) float v8f;

#define N_IN   2097152
#define KOFF   8
#define MPER   262144          // rulebook pairs per offset
#define N_OUT  524288
#define CIN    32
#define COUT   64
#define TILES_PER_K (MPER / 16)   // 16384
#define TOTAL_OUT  ((size_t)N_OUT * COUT)   // 33554432 floats

// ---------------- kernel 1: zero accumulator + stats ----------------
__global__ void sc_zero_kernel(float* __restrict__ out, float* __restrict__ stats) {
    size_t tid = (size_t)blockIdx.x * blockDim.x + threadIdx.x;
    size_t nthreads = (size_t)gridDim.x * blockDim.x;
    float4* o4 = (float4*)out;
    const size_t n4 = TOTAL_OUT / 4;
    for (size_t i = tid; i < n4; i += nthreads)
        o4[i] = make_float4(0.f, 0.f, 0.f, 0.f);
    if (blockIdx.x == 0 && threadIdx.x < 128)
        stats[threadIdx.x] = 0.f;
}

// ---------------- kernel 2: gather -> WMMA GEMM -> scatter-add ----------------
// One wave (32 threads) per block. blockIdx.y = kernel offset k.
// Rulebook indices are int64 but their values fit in 32 bits (N_IN=2M, N_OUT=512K),
// so we read only the low dword of each entry (little-endian) and keep all
// address arithmetic in 32-bit for SADDR+voffset global addressing.
__global__ void __launch_bounds__(32)
sc_conv_kernel(const float* __restrict__ feats,
               const float* __restrict__ W,
               const int* __restrict__ in_idx32,    // int64 viewed as int pairs
               const int* __restrict__ out_idx32,   // int64 viewed as int pairs
               float* __restrict__ out) {
    const int lane = threadIdx.x;      // 0..31, wave32
    const int half = lane >> 4;        // 0: lanes 0-15, 1: lanes 16-31
    const int l15  = lane & 15;
    const int k    = blockIdx.y;

    // ---- load B fragments for W[k] (32x64), kept in VGPRs for all tiles ----
    // V_WMMA_F32_16X16X4_F32 B layout (4x16): VGPR0 = rows {K=0 | K=2},
    // VGPR1 = rows {K=1 | K=3}, N striped over lanes within each half.
    const float* __restrict__ Wk = W + (size_t)k * (CIN * COUT);
    v2f bfrag[8][4];
#pragma unroll
    for (int kk = 0; kk < 8; ++kk) {
        const int r0 = kk * 4 + half * 2;
#pragma unroll
        for (int n = 0; n < 4; ++n) {
            const int col = n * 16 + l15;
            v2f b;
            b.x = Wk[(r0 + 0) * COUT + col];
            b.y = Wk[(r0 + 1) * COUT + col];
            bfrag[kk][n] = b;
        }
    }

    // low-dword view of this offset's rulebook rows (stride 2 ints per entry)
    const int* __restrict__ inI  = in_idx32  + (size_t)k * MPER * 2;
    const int* __restrict__ outI = out_idx32 + (size_t)k * MPER * 2;

    for (int tile = blockIdx.x; tile < TILES_PER_K; tile += gridDim.x) {
        const int base = tile * 16;

        // Each half of the wave loads the same 16 gather rows (low dword only).
        const unsigned row = (unsigned)inI[2 * (base + l15)];
        const float* __restrict__ arow = feats + row * CIN + half * 2;

        v8f acc0 = {}, acc1 = {}, acc2 = {}, acc3 = {};
#pragma unroll
        for (int kk = 0; kk < 8; ++kk) {
            // A layout (16x4): lane<16 holds K={0,1} of row M=lane,
            // lane>=16 holds K={2,3}; advanced by 4 per kk step.
            v2f a;
            a.x = arow[kk * 4 + 0];
            a.y = arow[kk * 4 + 1];
            acc0 = __builtin_amdgcn_wmma_f32_16x16x4_f32(false, a, false, bfrag[kk][0],
                                                         (short)0, acc0, false, false);
            acc1 = __builtin_amdgcn_wmma_f32_16x16x4_f32(false, a, false, bfrag[kk][1],
                                                         (short)0, acc1, false, false);
            acc2 = __builtin_amdgcn_wmma_f32_16x16x4_f32(false, a, false, bfrag[kk][2],
                                                         (short)0, acc2, false, false);
            acc3 = __builtin_amdgcn_wmma_f32_16x16x4_f32(false, a, false, bfrag[kk][3],
                                                         (short)0, acc3, false, false);
        }

        // ---- scatter-add: C/D layout VGPR j => M = j + half*8, N = l15 ----
#pragma unroll
        for (int j = 0; j < 8; ++j) {
            const int m = j + half * 8;
            const unsigned orow = (unsigned)outI[2 * (base + m)];
            float* op = out + orow * COUT + l15;
            unsafeAtomicAdd(op +  0, acc0[j]);
            unsafeAtomicAdd(op + 16, acc1[j]);
            unsafeAtomicAdd(op + 32, acc2[j]);
            unsafeAtomicAdd(op + 48, acc3[j]);
        }
    }
}

// ---------------- kernel 3: per-channel sum / sumsq ----------------
__global__ void sc_bn_stats_kernel(const float* __restrict__ out,
                                   float* __restrict__ stats) {
    __shared__ float bs[COUT];
    __shared__ float bss[COUT];
    if (threadIdx.x < COUT) { bs[threadIdx.x] = 0.f; bss[threadIdx.x] = 0.f; }
    __syncthreads();

    const int c = threadIdx.x & (COUT - 1);
    const int rowStart  = (int)((blockIdx.x * blockDim.x + threadIdx.x) >> 6);
    const int rowStride = (int)((gridDim.x * blockDim.x) >> 6);
    float s = 0.f, ss = 0.f;
    for (int r = rowStart; r < N_OUT; r += rowStride) {
        const float v = out[(size_t)r * COUT + c];
        s += v;
        ss += v * v;
    }
    atomicAdd(&bs[c], s);
    atomicAdd(&bss[c], ss);
    __syncthreads();
    if (threadIdx.x < COUT) {
        unsafeAtomicAdd(&stats[threadIdx.x], bs[threadIdx.x]);
        unsafeAtomicAdd(&stats[COUT + threadIdx.x], bss[threadIdx.x]);
    }
}

// ---------------- kernel 4: BN normalize + leaky relu (in place) ----------------
__global__ void sc_bn_apply_kernel(float* __restrict__ out,
                                   const float* __restrict__ stats,
                                   const float* __restrict__ gamma,
                                   const float* __restrict__ beta) {
    const float invN = 1.0f / (float)N_OUT;
    size_t tid = (size_t)blockIdx.x * blockDim.x + threadIdx.x;
    size_t nthreads = (size_t)gridDim.x * blockDim.x;
    for (size_t i = tid; i < TOTAL_OUT; i += nthreads) {
        const int c = (int)(i & (COUT - 1));
        const float mean  = stats[c] * invN;
        const float var   = stats[COUT + c] * invN - mean * mean;
        const float scale = rsqrtf(var + 1e-4f) * gamma[c];
        const float h = (out[i] - mean) * scale + beta[c];
        out[i] = h > 0.f ? h : h * (1.0f / 3.0f);
    }
}

extern "C" void kernel_launch(void* const* d_in, const int* in_sizes, int n_in,
                              void* d_out, int out_size, void* d_ws, size_t ws_size,
                              hipStream_t stream) {
    const float* feats    = (const float*)d_in[0];
    const float* W        = (const float*)d_in[1];
    const float* gamma    = (const float*)d_in[2];
    const float* beta     = (const float*)d_in[3];
    const int*   in_idx32  = (const int*)d_in[4];   // int64 rulebook, low dwords used
    const int*   out_idx32 = (const int*)d_in[5];
    float* out   = (float*)d_out;
    float* stats = (float*)d_ws;   // [0..63] sum, [64..127] sumsq

    sc_zero_kernel<<<dim3(8192), dim3(256), 0, stream>>>(out, stats);
    sc_conv_kernel<<<dim3(1024, KOFF), dim3(32), 0, stream>>>(feats, W, in_idx32, out_idx32, out);
    sc_bn_stats_kernel<<<dim3(1024), dim3(256), 0, stream>>>(out, stats);
    sc_bn_apply_kernel<<<dim3(4096), dim3(256), 0, stream>>>(out, stats, gamma, beta);
}